// ColorfulLoss_88510686036016
// MI455X (gfx1250) — compile-verified
//
#include <hip/hip_runtime.h>
#include <math.h>

typedef __attribute__((ext_vector_type(2))) float v2f;
typedef __attribute__((ext_vector_type(8))) float v8f;

// ---- problem shape (reference setup_inputs) ----
#define BB        16
#define HIN       256
#define WIN       256
#define HOUT      64
#define WOUT      64
#define NPIX      (BB * HOUT * WOUT)   // 65536
#define PLANE     (HOUT * WOUT)        // 4096
#define INV_2SIG2 (1.0f / 50.0f)       // 1/(2*5^2)

#define THREADS   256
#define NWAVES    (THREADS / 32)
#define TILE_STR  17                   // padded LDS row stride (coprime with 64 banks)
#define QMAX_PAD  512                  // compile-time bound on padded Q (actual Q ~313)

__global__ __launch_bounds__(THREADS)
void colorful_loss_main(const float* __restrict__ Zhat,
                        const float* __restrict__ ab_gt,
                        const float* __restrict__ class_weights,
                        const float* __restrict__ centres,
                        int Q,
                        float* __restrict__ partials)
{
    __shared__ float tile[NWAVES][32 * TILE_STR]; // per-wave 32px x 16 centre panel
    __shared__ float sBa[QMAX_PAD];               // -2*a_c
    __shared__ float sBb[QMAX_PAD];               // -2*b_c
    __shared__ float sBn[QMAX_PAD];               // |c|^2 (1e30 pad)
    __shared__ float red[THREADS];

    const int tid  = threadIdx.x;
    const int lane = tid & 31;
    const int wave = tid >> 5;
    const int p    = blockIdx.x * THREADS + tid;  // global pixel id (grid exact)
    const int b    = p >> 12;                     // /4096
    const int rem  = p & 4095;
    const int y    = rem >> 6;
    const int x    = rem & 63;

    const int qTiles = (Q + 15) >> 4;
    const int Qpad   = qTiles << 4;

    // ---- cooperative precompute of WMMA B-operand triplets into LDS (once/block) ----
    for (int i = tid; i < Qpad; i += THREADS) {
        const int cc = (i < Q) ? i : (Q - 1);
        const float ca = centres[2 * cc];
        const float cb = centres[2 * cc + 1];
        sBa[i] = -2.0f * ca;
        sBb[i] = -2.0f * cb;
        sBn[i] = (i < Q) ? (ca * ca + cb * cb) : 1.0e30f;  // pad -> never selected
    }

    // ---- 4x bilinear downsample: jax resize samples at 4*out+1.5 ->
    //      exact 0.25 average of input pixels (+1,+2) in each dim ----
    const size_t baseA = (size_t)b * (2 * HIN * WIN) + (size_t)(4 * y + 1) * WIN + (4 * x + 1);
    const size_t baseB = baseA + (size_t)HIN * WIN;
    const float av = 0.25f * (ab_gt[baseA] + ab_gt[baseA + 1] +
                              ab_gt[baseA + WIN] + ab_gt[baseA + WIN + 1]);
    const float bv = 0.25f * (ab_gt[baseB] + ab_gt[baseB + 1] +
                              ab_gt[baseB + WIN] + ab_gt[baseB + WIN + 1]);

    __syncthreads();   // sB* ready

    // ---- build A operands for V_WMMA_F32_16X16X4_F32 ----
    // dist'(p,c) = -2*a_p*a_c - 2*b_p*b_c + |c|^2  (|p|^2 cancels in top-k AND in
    // normalized soft weights): A row m = (a, b, 1, 0); B col n = (-2a_c, -2b_c, |c|^2, 0).
    // A 16x4 f32 layout: lanes 0-15 hold K=0,1 (VGPR0,1); lanes 16-31 hold K=2,3.
    const int half = lane >> 4;
    const int ncol = lane & 15;
    const float aUp = __shfl(av, 16 + ncol);      // pixel (lane+16)'s a, for A_hi
    const float bUp = __shfl(bv, 16 + ncol);
    v2f Alo, Ahi;
    Alo[0] = half ? 1.0f : av;   Alo[1] = half ? 0.0f : bv;
    Ahi[0] = half ? 1.0f : aUp;  Ahi[1] = half ? 0.0f : bUp;

    // ---- top-5 smallest distances per pixel (sorted ascending in registers) ----
    float d0 = 3e38f, d1 = 3e38f, d2 = 3e38f, d3 = 3e38f, d4 = 3e38f;
    int   i0 = 0,     i1 = 0,     i2 = 0,     i3 = 0,     i4 = 0;

    float* __restrict__ myTile = &tile[wave][0];
    const float* bRow0 = half ? sBn : sBa;        // K=2 row : K=0 row

    for (int j = 0; j < qTiles; ++j) {
        // B 4x16 layout mirrors A: lanes 0-15 hold rows K=0,1; lanes 16-31 rows K=2,3.
        const int cidx = (j << 4) + ncol;
        v2f Bt;
        Bt[0] = bRow0[cidx];                       // -2a_c (half0) | |c|^2 (half1)
        const float bbv = sBb[cidx];               // unconditional load, then select
        Bt[1] = half ? 0.0f : bbv;                 // -2b_c (half0) | 0     (half1)

        v8f acc = {};
        v8f Dlo = __builtin_amdgcn_wmma_f32_16x16x4_f32(false, Alo, false, Bt,
                                                        (short)0, acc, false, false);
        v8f Dhi = __builtin_amdgcn_wmma_f32_16x16x4_f32(false, Ahi, false, Bt,
                                                        (short)0, acc, false, false);

        // D layout: VGPR r = row (r + 8*half), col = lane&15. Spill to wave-private
        // LDS (stride 17 -> conflict-free row reads) to transpose to per-lane rows.
        #pragma unroll
        for (int r = 0; r < 8; ++r) {
            const int m = r + (half << 3);
            myTile[m * TILE_STR + ncol]        = Dlo[r];
            myTile[(16 + m) * TILE_STR + ncol] = Dhi[r];
        }

        // batch-load my pixel row's 16 fresh distances (merged ds loads, one wait)
        float vv[16];
        #pragma unroll
        for (int t = 0; t < 16; ++t) vv[t] = myTile[lane * TILE_STR + t];

        // single-level guard + branchless sorted insert (strict '<' keeps jax's
        // tie-to-lower-index order since q scans ascending)
        #pragma unroll
        for (int t = 0; t < 16; ++t) {
            const float v  = vv[t];
            const int   ci = (j << 4) + t;
            if (v < d4) {
                const bool c0 = v < d0, c1 = v < d1, c2 = v < d2, c3 = v < d3;
                const float nd0 = c0 ? v  : d0;            const int ni0 = c0 ? ci : i0;
                const float nd1 = c0 ? d0 : (c1 ? v : d1); const int ni1 = c0 ? i0 : (c1 ? ci : i1);
                const float nd2 = c1 ? d1 : (c2 ? v : d2); const int ni2 = c1 ? i1 : (c2 ? ci : i2);
                const float nd3 = c2 ? d2 : (c3 ? v : d3); const int ni3 = c2 ? i2 : (c3 ? ci : i3);
                const float nd4 = c3 ? d3 : v;             const int ni4 = c3 ? i3 : ci;
                d0 = nd0; d1 = nd1; d2 = nd2; d3 = nd3; d4 = nd4;
                i0 = ni0; i1 = ni1; i2 = ni2; i3 = ni3; i4 = ni4;
            }
        }
    }

    // ---- two-pass logsumexp over Q logits (lane = x -> coalesced 128B/wave/q).
    //      Zhat (80 MB) is L2-resident (192 MB L2): pass 2 re-reads hit L2. ----
    const float* __restrict__ zp = Zhat + (size_t)b * Q * PLANE + (size_t)y * WOUT + x;

    float m0 = -3.0e38f, m1 = -3.0e38f, m2 = -3.0e38f, m3 = -3.0e38f;
    int q = 0;
    for (; q + 4 <= Q; q += 4) {
        const int qp = (q + 32 < Q) ? (q + 32) : (Q - 1);
        __builtin_prefetch(zp + (size_t)qp * PLANE, 0, 0);   // stream 32 planes ahead
        const float z0 = zp[(size_t)(q + 0) * PLANE];
        const float z1 = zp[(size_t)(q + 1) * PLANE];
        const float z2 = zp[(size_t)(q + 2) * PLANE];
        const float z3 = zp[(size_t)(q + 3) * PLANE];
        m0 = fmaxf(m0, z0); m1 = fmaxf(m1, z1);
        m2 = fmaxf(m2, z2); m3 = fmaxf(m3, z3);
    }
    for (; q < Q; ++q) m0 = fmaxf(m0, zp[(size_t)q * PLANE]);
    const float mx = fmaxf(fmaxf(m0, m1), fmaxf(m2, m3));

    float s0 = 0.0f, s1 = 0.0f, s2 = 0.0f, s3 = 0.0f;
    q = 0;
    for (; q + 4 <= Q; q += 4) {
        const float z0 = zp[(size_t)(q + 0) * PLANE];
        const float z1 = zp[(size_t)(q + 1) * PLANE];
        const float z2 = zp[(size_t)(q + 2) * PLANE];
        const float z3 = zp[(size_t)(q + 3) * PLANE];
        s0 += __expf(z0 - mx); s1 += __expf(z1 - mx);
        s2 += __expf(z2 - mx); s3 += __expf(z3 - mx);
    }
    for (; q < Q; ++q) s0 += __expf(zp[(size_t)q * PLANE] - mx);
    const float lse = mx + __logf((s0 + s1) + (s2 + s3));

    // ---- gather top-5 logits (L2-resident), soft weights, CE ----
    const float z0 = zp[(size_t)i0 * PLANE];
    const float z1 = zp[(size_t)i1 * PLANE];
    const float z2 = zp[(size_t)i2 * PLANE];
    const float z3 = zp[(size_t)i3 * PLANE];
    const float z4 = zp[(size_t)i4 * PLANE];
    const float w0 = 1.0f;  // exp((d0-d0)/2s^2); |p|^2 shift cancels in normalization
    const float w1 = __expf((d0 - d1) * INV_2SIG2);
    const float w2 = __expf((d0 - d2) * INV_2SIG2);
    const float w3 = __expf((d0 - d3) * INV_2SIG2);
    const float w4 = __expf((d0 - d4) * INV_2SIG2);
    const float wsum = w0 + w1 + w2 + w3 + w4;
    const float dot  = w0 * z0 + w1 * z1 + w2 * z2 + w3 * z3 + w4 * z4;
    const float ce   = lse - dot / wsum;       // -sum(w*(z-lse)), sum(w)=1
    const float contrib = class_weights[i0] * ce;

    // ---- deterministic block tree reduction -> one partial per block ----
    red[tid] = contrib;
    __syncthreads();
    #pragma unroll
    for (int s = THREADS / 2; s > 0; s >>= 1) {
        if (tid < s) red[tid] += red[tid + s];
        __syncthreads();
    }
    if (tid == 0) partials[blockIdx.x] = red[0];
}

__global__ __launch_bounds__(256)
void colorful_loss_finalize(const float* __restrict__ partials, int nparts,
                            float* __restrict__ out, float invN)
{
    __shared__ float red[256];
    float s = 0.0f;
    for (int i = threadIdx.x; i < nparts; i += 256) s += partials[i];
    red[threadIdx.x] = s;
    __syncthreads();
    #pragma unroll
    for (int st = 128; st > 0; st >>= 1) {
        if (threadIdx.x < st) red[threadIdx.x] += red[threadIdx.x + st];
        __syncthreads();
    }
    if (threadIdx.x == 0) out[0] = red[0] * invN;
}

extern "C" void kernel_launch(void* const* d_in, const int* in_sizes, int n_in,
                              void* d_out, int out_size, void* d_ws, size_t ws_size,
                              hipStream_t stream)
{
    const float* Zhat    = (const float*)d_in[0];
    const float* ab_gt   = (const float*)d_in[1];
    const float* cw      = (const float*)d_in[2];
    const float* centres = (const float*)d_in[3];
    const int Q = in_sizes[2];                 // class_weights length == Q

    float* out      = (float*)d_out;
    float* partials = (float*)d_ws;            // nblocks floats of scratch

    const int nblocks = NPIX / THREADS;        // 256
    colorful_loss_main<<<nblocks, THREADS, 0, stream>>>(Zhat, ab_gt, cw, centres, Q, partials);
    colorful_loss_finalize<<<1, 256, 0, stream>>>(partials, nblocks, out, 1.0f / (float)NPIX);
}